// RNNSequence_5669356836206
// MI455X (gfx1250) — compile-verified
//
#include <hip/hip_runtime.h>
#include <math.h>

// CDNA5 / gfx1250. wave32.
typedef __attribute__((ext_vector_type(2))) float v2f;
typedef __attribute__((ext_vector_type(8))) float v8f;

#define BATCH 32
#define SEQ   2048
#define DIN   512
#define DH    512
#define MTOT  (BATCH * SEQ)   // 65536 rows of the input-projection GEMM

// ---------------------------------------------------------------------------
// Phase 1: out[m, n] = sum_k x[m, k] * W_ih[k, n] + b[n]
// One block = 128x64 output tile. 8 waves; wave w does rows [m0+16w, +16),
// all 64 columns of the block's N-stripe (4 accumulators of 16x16).
// Uses V_WMMA_F32_16X16X4_F32 (native fp32 matrix op -> no precision loss).
// ---------------------------------------------------------------------------
__global__ __launch_bounds__(256) void xw_gemm_f32_wmma(
    const float* __restrict__ X,     // [MTOT, DIN]
    const float* __restrict__ W,     // [DIN, DH]
    const float* __restrict__ bias,  // [DH]
    float* __restrict__ out)         // [MTOT, DH]
{
    const int lane = threadIdx.x & 31;
    const int wave = threadIdx.x >> 5;
    const int n0   = blockIdx.x * 64;              // N-stripe (gridDim.x = DH/64 = 8)
    const int m0   = blockIdx.y * 128 + wave * 16; // M tile   (gridDim.y = MTOT/128 = 512)

    const int mrow  = lane & 15;        // M row (A) and N col (B, C/D) within tile
    const int khalf = (lane >> 4) << 1; // 0 for lanes 0-15, 2 for lanes 16-31

    v8f acc0 = {0.f,0.f,0.f,0.f,0.f,0.f,0.f,0.f};
    v8f acc1 = acc0, acc2 = acc0, acc3 = acc0;

    const float* xrow = X + (size_t)(m0 + mrow) * DIN + khalf;

    for (int k = 0; k < DIN; k += 4) {
        // A tile 16x4: lane holds X[m0+mrow][k+khalf .. +1] (contiguous 8B)
        v2f a;
        a.x = xrow[k + 0];
        a.y = xrow[k + 1];

        const float* wr0 = W + (size_t)(k + khalf)     * DH;
        const float* wr1 = W + (size_t)(k + khalf + 1) * DH;

        // B tiles 4x16 per 16-col subtile: lane holds W[k+khalf][n], W[k+khalf+1][n]
        v2f b0; b0.x = wr0[n0 +  0 + mrow]; b0.y = wr1[n0 +  0 + mrow];
        v2f b1; b1.x = wr0[n0 + 16 + mrow]; b1.y = wr1[n0 + 16 + mrow];
        v2f b2; b2.x = wr0[n0 + 32 + mrow]; b2.y = wr1[n0 + 32 + mrow];
        v2f b3; b3.x = wr0[n0 + 48 + mrow]; b3.y = wr1[n0 + 48 + mrow];

        acc0 = __builtin_amdgcn_wmma_f32_16x16x4_f32(false, a, false, b0, (short)0, acc0, false, false);
        acc1 = __builtin_amdgcn_wmma_f32_16x16x4_f32(false, a, false, b1, (short)0, acc1, false, false);
        acc2 = __builtin_amdgcn_wmma_f32_16x16x4_f32(false, a, false, b2, (short)0, acc2, false, false);
        acc3 = __builtin_amdgcn_wmma_f32_16x16x4_f32(false, a, false, b3, (short)0, acc3, false, false);
    }

    // C/D layout: VGPR r -> (M = r + 8*(lane>=16), N = lane&15)
    const int rowoff = (lane >> 4) * 8;
    v8f accs[4] = {acc0, acc1, acc2, acc3};
    #pragma unroll
    for (int s = 0; s < 4; ++s) {
        const int n  = n0 + s * 16 + mrow;
        const float bb = bias[n];
        #pragma unroll
        for (int r = 0; r < 8; ++r) {
            out[(size_t)(m0 + rowoff + r) * DH + n] = accs[s][r] + bb;
        }
    }
}

// ---------------------------------------------------------------------------
// Phase 2: strictly-sequential scan  h_t = tanh(xw_t + h_{t-1} @ W_hh).
// Batch rows are independent -> one workgroup (512 threads = 16 wave32) per
// batch row; h lives in LDS (2 KB), W_hh (1 MB) streams from L2 each step
// (coalesced: at each k, the wave reads W_hh[k][j..j+31], 128B contiguous).
// In-place over d_out: reads xw_t then overwrites with h_t.
// ---------------------------------------------------------------------------
__global__ __launch_bounds__(512) void rnn_scan(
    const float* __restrict__ W,   // W_hh [DH, DH]
    float* __restrict__ out)       // [BATCH, SEQ, DH]; xw in, h out
{
    __shared__ float h[DH];
    const int j = threadIdx.x;     // output element owned by this thread
    const int b = blockIdx.x;      // batch row

    h[j] = 0.0f;
    __syncthreads();

    float* row = out + (size_t)b * SEQ * DH;

    for (int t = 0; t < SEQ; ++t) {
        float acc = row[(size_t)t * DH + j];

        #pragma unroll 4
        for (int k = 0; k < DH; k += 4) {
            const float4 hv = *(const float4*)&h[k];
            acc = fmaf(hv.x, W[(size_t)(k + 0) * DH + j], acc);
            acc = fmaf(hv.y, W[(size_t)(k + 1) * DH + j], acc);
            acc = fmaf(hv.z, W[(size_t)(k + 2) * DH + j], acc);
            acc = fmaf(hv.w, W[(size_t)(k + 3) * DH + j], acc);
        }

        const float hn = tanhf(acc);

        __syncthreads();          // everyone done reading h_{t-1}
        h[j] = hn;
        row[(size_t)t * DH + j] = hn;
        __syncthreads();          // h_t visible before next step reads
    }
}

// ---------------------------------------------------------------------------
extern "C" void kernel_launch(void* const* d_in, const int* in_sizes, int n_in,
                              void* d_out, int out_size, void* d_ws, size_t ws_size,
                              hipStream_t stream) {
    const float* x    = (const float*)d_in[0];  // [32, 2048, 512]
    const float* W_ih = (const float*)d_in[1];  // [512, 512]
    const float* W_hh = (const float*)d_in[2];  // [512, 512]
    const float* bias = (const float*)d_in[3];  // [512]
    float* out = (float*)d_out;                 // [32, 2048, 512]

    dim3 g1(DH / 64, MTOT / 128);               // (8, 512)
    xw_gemm_f32_wmma<<<g1, 256, 0, stream>>>(x, W_ih, bias, out);

    rnn_scan<<<BATCH, DH, 0, stream>>>(W_hh, out);
}